// LIFAct_2628519985323
// MI455X (gfx1250) — compile-verified
//
#include <hip/hip_runtime.h>
#include <stdint.h>

// LIF scan over T=4 on [T,B,C,H,W] f32. Pure streaming kernel:
// 268 MB of traffic @ 23.3 TB/s HBM => ~11.5 us floor. Strategy:
//  - one float4 lane-column per thread, all 4 time-plane loads issued
//    up front (independent of the sequential membrane recurrence)
//  - 128-bit non-temporal loads/stores (stream > 192MB L2, zero reuse)
//  - wave32-friendly 256-thread blocks (8 waves)

typedef float v4f __attribute__((ext_vector_type(4)));

#define LIF_T 4
#define LIF_DECAY 0.25f

__device__ __forceinline__ float lif_step(float& m, float xv, float fr) {
    m = __builtin_fmaf(m, LIF_DECAY, xv);          // mem = mem*decay + x
    const bool fired = __builtin_fabsf(m) >= 0.5f; // |mem/V_TH| >= 0.5
    const float s = fired ? __builtin_copysignf(1.0f, m) : 0.0f;
    m = fired ? 0.0f : m;                          // reset where fired
    return s * fr;                                 // spike * fire_ratio
}

__global__ __launch_bounds__(256) void lif_scan_kernel(
    const v4f* __restrict__ x,    // [T][n4] float4 view
    const float* __restrict__ fire_ratio_p,
    v4f* __restrict__ out,        // [T][n4] float4 view
    int n4)                       // float4 elements per time plane
{
    const int i = blockIdx.x * blockDim.x + threadIdx.x;
    if (i >= n4) return;

    const float fr = fire_ratio_p[0];   // uniform -> scalar load

    // Issue all T independent plane loads before the recurrence so the
    // memory pipe stays full; each is a global_load_b128 th:NT.
    v4f xt[LIF_T];
#pragma unroll
    for (int t = 0; t < LIF_T; ++t)
        xt[t] = __builtin_nontemporal_load(x + (size_t)t * n4 + i);

    // Membrane state as plain scalars (mutable refs to vector elements
    // are not allowed on ext_vector types).
    float m[4] = {0.0f, 0.0f, 0.0f, 0.0f};

#pragma unroll
    for (int t = 0; t < LIF_T; ++t) {
        v4f o;
#pragma unroll
        for (int k = 0; k < 4; ++k) {
            const float xv = xt[t][k];          // rvalue read: OK
            o[k] = lif_step(m[k], xv, fr);
        }
        __builtin_nontemporal_store(o, out + (size_t)t * n4 + i);
    }
}

extern "C" void kernel_launch(void* const* d_in, const int* in_sizes, int n_in,
                              void* d_out, int out_size, void* d_ws, size_t ws_size,
                              hipStream_t stream) {
    const float* x  = (const float*)d_in[0];       // [T,B,C,H,W] f32
    const float* fr = (const float*)d_in[1];       // scalar fire_ratio
    float* out      = (float*)d_out;               // [T,B,C,H,W] f32

    const long long total = (long long)in_sizes[0];   // T*B*C*H*W
    const int plane = (int)(total / LIF_T);           // elements per time plane
    const int n4    = plane / 4;                      // float4 per plane

    const int block = 256;                            // 8 wave32 waves
    const int grid  = (n4 + block - 1) / block;

    lif_scan_kernel<<<grid, block, 0, stream>>>(
        (const v4f*)x, fr, (v4f*)out, n4);
}